// CNOLReLu_12936441495785
// MI455X (gfx1250) — compile-verified
//
#include <hip/hip_runtime.h>
#include <hip/hip_bf16.h>

typedef __attribute__((ext_vector_type(2))) float v2f;
typedef __attribute__((ext_vector_type(8))) float v8f;

#define WMMA_F32X4(a, b, c) \
  __builtin_amdgcn_wmma_f32_16x16x4_f32(false, (a), false, (b), (short)0, (c), false, false)

#define NEG_SLOPE 0.01f
#define PAD 4               // zero columns on each side of weight band
#define LDU (128 + 2 * PAD) // 136, padded row stride of U (256x128)
#define LDD (256 + 2 * PAD) // 264, padded row stride of D (128x256)

// Keys cubic, a = -0.5 (jax.image bicubic kernel)
__device__ __forceinline__ float keys_cubic(float x) {
  x = fabsf(x);
  float out = ((1.5f * x - 2.5f) * x) * x + 1.0f;
  if (x >= 1.0f) out = ((-0.5f * x + 2.5f) * x - 4.0f) * x + 2.0f;
  if (x >= 2.0f) out = 0.0f;
  return out;
}

// Block 0: Upad (256 x LDU) upsample weights, col j <-> tap k = j - PAD.
// Block 1: Dpad (128 x LDD) antialiased downsample weights.
// Column-normalized exactly like jax.image compute_weight_mat; pad cols are 0.
__global__ void init_weights_kernel(float* __restrict__ U, float* __restrict__ Dm) {
  int t = threadIdx.x;
  if (blockIdx.x == 0) {
    int p = t;  // 256 threads
    float src = 0.5f * (float)p - 0.25f;  // (p+0.5)/2 - 0.5
    float s = 0.0f;
    for (int h = 0; h < 128; ++h) s += keys_cubic(src - (float)h);
    float inv = (fabsf(s) > 1e-6f) ? (1.0f / s) : 0.0f;
    for (int j = 0; j < LDU; ++j) {
      int h = j - PAD;
      U[p * LDU + j] =
          (h >= 0 && h < 128) ? keys_cubic(src - (float)h) * inv : 0.0f;
    }
  } else if (t < 128) {
    int i = t;
    float src = 2.0f * (float)i + 0.5f;  // (i+0.5)*2 - 0.5
    float s = 0.0f;
    for (int q = 0; q < 256; ++q) s += keys_cubic((src - (float)q) * 0.5f);
    float inv = (fabsf(s) > 1e-6f) ? (1.0f / s) : 0.0f;
    for (int j = 0; j < LDD; ++j) {
      int q = j - PAD;
      Dm[i * LDD + j] =
          (q >= 0 && q < 256) ? keys_cubic((src - (float)q) * 0.5f) * inv : 0.0f;
    }
  }
}

// A-fragment from padded weights: one aligned 8B load per lane, never OOB.
__device__ __forceinline__ v2f load_w_frag(const float* __restrict__ W, int ldpad,
                                           int m0, int kbase) {
  int lane = threadIdx.x & 31;
  int m = m0 + (lane & 15);
  int kk = kbase + PAD + ((lane >> 4) << 1);  // padded col, even, in range
  return *(const v2f*)(W + m * ldpad + kk);
}

// B-fragment with clamped row index: out-of-band rows hit weight==0, so the
// clamped (garbage) value contributes nothing. Unconditional loads.
__device__ __forceinline__ v2f load_x_frag(const float* __restrict__ X, int ld,
                                           int kbase, int Klim, int n0) {
  int lane = threadIdx.x & 31;
  int n = n0 + (lane & 15);
  int kk = kbase + ((lane >> 4) << 1);
  int k0 = min(max(kk, 0), Klim - 1);
  int k1 = min(max(kk + 1, 0), Klim - 1);
  v2f b;
  b.x = X[k0 * ld + n];
  b.y = X[k1 * ld + n];
  return b;
}

// dst[b][m][0..N) = sum_k W[m][k] * src[b][k][0..N), banded:
// for m-block m0, only k in [m0*K/M - PAD, +4*KSTEPS) can be nonzero.
template <int M, int K, int KSTEPS, int LDW>
__global__ void __launch_bounds__(256)
band_matmul_kernel(const float* __restrict__ W, const float* __restrict__ src,
                   float* __restrict__ dst) {
  constexpr int N = 16384;
  int wave = threadIdx.x >> 5;
  int lane = threadIdx.x & 31;
  int n0 = blockIdx.x * 128 + wave * 16;
  int m0 = blockIdx.y * 16;
  long b = blockIdx.z;
  const float* S = src + b * (long)K * N;
  float* O = dst + b * (long)M * N;
  int kstart = (m0 * K) / M - PAD;

  v8f acc = {};
#pragma unroll
  for (int t = 0; t < KSTEPS; ++t) {
    int k = kstart + 4 * t;
    v2f a = load_w_frag(W, LDW, m0, k);
    v2f bf = load_x_frag(S, N, k, K, n0);
    acc = WMMA_F32X4(a, bf, acc);
  }
  int col = n0 + (lane & 15);
  int rbase = m0 + ((lane >> 4) << 3);
#pragma unroll
  for (int r = 0; r < 8; ++r) O[(long)(rbase + r) * N + col] = acc[r];
}

// Per (b,p) slice of T1 (128w x 128c), c-strip of 32:
//   F[q][c'] = leaky( sum_w U[q][w] * slice[w][cs+c'] )   (up-W + act, into LDS)
//   slice[j][cs+c'] = sum_q D[j][q] * F[q][c']            (down-W, in place)
__global__ void __launch_bounds__(256)
row_upactdown_kernel(const float* __restrict__ U, const float* __restrict__ Dm,
                     float* __restrict__ T1) {
  __shared__ float F[256][33];  // padded stride vs 64-bank LDS
  int wave = threadIdx.x >> 5;
  int lane = threadIdx.x & 31;
  int cs = blockIdx.x * 32;
  long sliceIdx = (long)blockIdx.z * 256 + blockIdx.y;
  float* slice = T1 + sliceIdx * 16384;

  // Phase A: 16 q-blocks x 2 c-tiles = 32 tiles over 8 waves
  for (int T = wave; T < 32; T += 8) {
    int qb = T >> 1, ct = T & 1;
    int q0 = qb * 16, n0 = ct * 16;
    int kstart = 8 * qb - PAD;
    v8f acc = {};
#pragma unroll
    for (int t = 0; t < 4; ++t) {
      int k = kstart + 4 * t;
      v2f a = load_w_frag(U, LDU, q0, k);
      v2f bf = load_x_frag(slice + cs, 128, k, 128, n0);
      acc = WMMA_F32X4(a, bf, acc);
    }
    int col = n0 + (lane & 15);
    int rbase = q0 + ((lane >> 4) << 3);
#pragma unroll
    for (int r = 0; r < 8; ++r) {
      float v = acc[r];
      F[rbase + r][col] = (v >= 0.0f) ? v : NEG_SLOPE * v;
    }
  }
  __syncthreads();

  // Phase B: 8 j-blocks x 2 c-tiles = 16 tiles over 8 waves
  for (int T = wave; T < 16; T += 8) {
    int jb = T >> 1, ct = T & 1;
    int j0 = jb * 16, n0 = ct * 16;
    int kstart = 32 * jb - PAD;
    v8f acc = {};
#pragma unroll
    for (int t = 0; t < 10; ++t) {
      int k = kstart + 4 * t;
      v2f a = load_w_frag(Dm, LDD, j0, k);
      int n = n0 + (lane & 15);
      int kk = k + ((lane >> 4) << 1);
      int k0 = min(max(kk, 0), 255);
      int k1 = min(max(kk + 1, 0), 255);
      v2f bf;
      bf.x = F[k0][n];
      bf.y = F[k1][n];
      acc = WMMA_F32X4(a, bf, acc);
    }
    int col = cs + n0 + (lane & 15);
    int rbase = j0 + ((lane >> 4) << 3);
#pragma unroll
    for (int r = 0; r < 8; ++r) slice[(rbase + r) * 128 + col] = acc[r];
  }
}

extern "C" void kernel_launch(void* const* d_in, const int* in_sizes, int n_in,
                              void* d_out, int out_size, void* d_ws, size_t ws_size,
                              hipStream_t stream) {
  const float* x = (const float*)d_in[0];  // (16,128,128,128) fp32 NHWC
  float* out = (float*)d_out;              // (16,128,128,128) fp32
  char* ws = (char*)d_ws;

  float* U  = (float*)ws;                           // 256*136 f32 = 136 KB
  float* Dm = (float*)(ws + 256 * LDU * 4);         // 128*264 f32 = 132 KB
  float* T1 = (float*)(ws + (512 << 10));           // 256 MiB at 512 KB offset

  init_weights_kernel<<<2, 256, 0, stream>>>(U, Dm);

  // Stage 1: up-H  T1[b][p][w*128+c] = sum_h U[p][h] * x[b][h][w*128+c]
  dim3 g1(16384 / 128, 256 / 16, 16);
  band_matmul_kernel<256, 128, 4, LDU><<<g1, 256, 0, stream>>>(U, x, T1);

  // Stage 2: per fine row: up-W + leaky + down-W, in place on T1
  dim3 g2(4, 256, 16);
  row_upactdown_kernel<<<g2, 256, 0, stream>>>(U, Dm, T1);

  // Stage 3: down-H  out[b][i][j*128+c] = sum_p D[i][p] * T1[b][p][j*128+c]
  dim3 g3(16384 / 128, 128 / 16, 16);
  band_matmul_kernel<128, 256, 10, LDD><<<g3, 256, 0, stream>>>(Dm, T1, out);
}